// scFuMamba_stage2_80865644249262
// MI455X (gfx1250) — compile-verified
//
#include <hip/hip_runtime.h>
#include <hip/hip_bf16.h>
#include <math.h>

// ---------------------------------------------------------------------------
// scFuMamba stage-2 forward for MI455X (gfx1250, wave32, WMMA)
//
// Pipeline (RNA):  tokens GEMM (HBM-bound, plain tiled) -> embed+pos+cls ->
//   in_proj via V_WMMA_F32_16X16X4_F32 -> depthwise causal conv+SiLU (in-place)
//   -> fused x_proj + dt_proj(softplus) -> selective scan (lane per (b,d),
//   16-state in VGPRs) emitting y only for tokens 1867..2000 -> gate ->
//   out_proj (WMMA) -> LN + segment max  => c1
// ADT path: one fused kernel per batch (L=15, all in LDS)  => c2
// Fusion: one block per batch => logits / final / o1 / o2
// ---------------------------------------------------------------------------

#define EMB     128
#define DIN     256
#define DST     16
#define LRNA    2001
#define SEG0    1867
#define SEGL    134
#define BATCH   32
#define NTOK    4000

typedef __attribute__((ext_vector_type(2))) float v2f;
typedef __attribute__((ext_vector_type(8))) float v8f;

__device__ __forceinline__ float sigf(float x) { return 1.f / (1.f + __expf(-x)); }
__device__ __forceinline__ float siluf(float x) { return x * sigf(x); }
__device__ __forceinline__ float softplusf(float x) {
  return (x > 20.f) ? x : log1pf(__expf(x));
}

// ---------------------------------------------------------------------------
// K1: t0[b,j] = sum_k patches1[b,k] * tokens_w[j,k] + tokens_b[j]
// 64 MB weight read -> pure HBM roofline; LDS-tiled, coalesced W loads.
// ---------------------------------------------------------------------------
__global__ void k_tokens_gemm(const float* __restrict__ P,
                              const float* __restrict__ W,
                              const float* __restrict__ bias,
                              float* __restrict__ out) {
  __shared__ float Wl[64][129];
  __shared__ float Pl[32][129];
  const int j0 = blockIdx.x * 64;
  const int tid = threadIdx.x;               // 256
  const int jl = tid & 63;
  const int bg = tid >> 6;                   // 4 groups of 8 batches
  float acc[8] = {0,0,0,0,0,0,0,0};
  for (int k0 = 0; k0 < NTOK; k0 += 128) {
    for (int i = tid; i < 64 * 128; i += 256) {
      int r = i >> 7, c = i & 127;
      int j = j0 + r, k = k0 + c;
      Wl[r][c] = (j < NTOK && k < NTOK) ? W[(long)j * NTOK + k] : 0.f;
    }
    for (int i = tid; i < 32 * 128; i += 256) {
      int r = i >> 7, c = i & 127;
      int k = k0 + c;
      Pl[r][c] = (k < NTOK) ? P[r * NTOK + k] : 0.f;
    }
    __syncthreads();
    for (int c = 0; c < 128; c++) {
      float w = Wl[jl][c];
#pragma unroll
      for (int i = 0; i < 8; i++) acc[i] += w * Pl[bg * 8 + i][c];
    }
    __syncthreads();
  }
  int j = j0 + jl;
  if (j < NTOK) {
    float bj = bias[j];
#pragma unroll
    for (int i = 0; i < 8; i++) out[(bg * 8 + i) * NTOK + j] = acc[i] + bj;
  }
}

// ---------------------------------------------------------------------------
// K2: build seq[b, 0..2000, 128] = [cls ; t0.reshape(2000,2)@emb_w.T + emb_b + pos]
// ---------------------------------------------------------------------------
__global__ void k_embed(const float* __restrict__ t0,
                        const float* __restrict__ emb_w,  // (128,2)
                        const float* __restrict__ emb_b,
                        const float* __restrict__ pos,    // (2000,128)
                        const float* __restrict__ cls,    // (128)
                        float* __restrict__ seq) {
  int c = blockIdx.x, b = blockIdx.y, e = threadIdx.x;
  float v;
  if (c == 0) {
    v = cls[e];
  } else {
    int cc = c - 1;
    v = t0[b * NTOK + 2 * cc] * emb_w[2 * e] +
        t0[b * NTOK + 2 * cc + 1] * emb_w[2 * e + 1] +
        emb_b[e] + pos[cc * EMB + e];
  }
  seq[((long)b * LRNA + c) * EMB + e] = v;
}

// ---------------------------------------------------------------------------
// K3: generic fp32 WMMA GEMM:  C[M,N] = A'[M,K] @ W[N,K]^T
// A row gather: arow(m) = (m / rowDiv) * rowMul + rowAdd + (m % rowDiv)
//   (identity when rowDiv==rowMul, rowAdd==0)
// Per-wave 16x16 tile, K stepped by 4 via V_WMMA_F32_16X16X4_F32.
// A 16x4 fragment: lanes 0-15 -> M rows, VGPR pair holds K={0,1} (lo half)
// and K={2,3} (hi half).  B 4x16 mirrored.  C/D: v8f per ISA layout.
// ---------------------------------------------------------------------------
__global__ void k_wmma_gemm(const float* __restrict__ A,
                            const float* __restrict__ W,
                            float* __restrict__ C,
                            int M, int N, int K,
                            int rowDiv, int rowMul, int rowAdd) {
  const int lane = threadIdx.x & 31;
  const int wv = threadIdx.x >> 5;
  const int mt = blockIdx.x * 4 + wv;
  if (mt * 16 >= M) return;
  const int nt = blockIdx.y;
  const int l16 = lane & 15;
  const int hi = lane >> 4;                 // 0: K 0/1, 1: K 2/3

  const int mrow = mt * 16 + l16;
  const long arow = (long)(mrow / rowDiv) * rowMul + rowAdd + (mrow % rowDiv);
  const float* Ap = A + arow * (long)K + (hi ? 2 : 0);
  const float* Wp = W + (long)(nt * 16 + l16) * K + (hi ? 2 : 0);

  v8f c = {0.f, 0.f, 0.f, 0.f, 0.f, 0.f, 0.f, 0.f};
#pragma unroll 4
  for (int k = 0; k < K; k += 4) {
    v2f a = *(const v2f*)(Ap + k);
    v2f bfrag = *(const v2f*)(Wp + k);
    c = __builtin_amdgcn_wmma_f32_16x16x4_f32(
        /*neg_a=*/false, a, /*neg_b=*/false, bfrag,
        /*c_mod=*/(short)0, c, /*reuse_a=*/false, /*reuse_b=*/false);
  }
  const int n = nt * 16 + l16;
  const long mbase = (long)mt * 16 + (hi ? 8 : 0);
#pragma unroll
  for (int r = 0; r < 8; r++) C[(mbase + r) * (long)N + n] = c[r];
}

// ---------------------------------------------------------------------------
// K4: depthwise causal conv(k=4) + bias + SiLU, in-place on xi -> xa.
// One thread per (b,d) column; window kept in registers so the in-place
// update is race-free (column is thread-private).
// ---------------------------------------------------------------------------
__global__ void k_conv_silu(float* __restrict__ xa,
                            const float* __restrict__ cw,  // (256,1,4)
                            const float* __restrict__ cb) {
  const int b = blockIdx.x;     // 32
  const int d = threadIdx.x;    // 256
  const float w0 = cw[d * 4 + 0], w1 = cw[d * 4 + 1],
              w2 = cw[d * 4 + 2], w3 = cw[d * 4 + 3];
  const float bb = cb[d];
  float x0 = 0.f, x1 = 0.f, x2 = 0.f, x3 = 0.f;
  for (int t = 0; t < LRNA; t++) {
    long idx = ((long)b * LRNA + t) * DIN + d;
    x0 = x1; x1 = x2; x2 = x3; x3 = xa[idx];
    float v = w0 * x0 + w1 * x1 + w2 * x2 + w3 * x3 + bb;
    xa[idx] = siluf(v);
  }
}

// ---------------------------------------------------------------------------
// K5: fused x_proj (40 outs) + dt_proj + softplus.  One block per (b,t) row.
// ---------------------------------------------------------------------------
__global__ void k_xproj_dt(const float* __restrict__ xa,
                           const float* __restrict__ xw,   // (40,256)
                           const float* __restrict__ dtw,  // (256,8)
                           const float* __restrict__ dtbias,
                           float* __restrict__ Bb, float* __restrict__ Cb,
                           float* __restrict__ dtout) {
  __shared__ float row[DIN];
  __shared__ float dbl[40];
  const long m = blockIdx.x;
  const int tid = threadIdx.x;  // 256
  row[tid] = xa[m * DIN + tid];
  __syncthreads();
  if (tid < 40) {
    float a = 0.f;
    const float* w = xw + tid * DIN;
    for (int k = 0; k < DIN; k++) a += row[k] * w[k];
    dbl[tid] = a;
    if (tid >= 8 && tid < 24)      Bb[m * DST + (tid - 8)] = a;
    else if (tid >= 24)            Cb[m * DST + (tid - 24)] = a;
  }
  __syncthreads();
  float s = dtbias[tid];
#pragma unroll
  for (int r = 0; r < 8; r++) s += dbl[r] * dtw[tid * 8 + r];
  dtout[m * DIN + tid] = softplusf(s);
}

// ---------------------------------------------------------------------------
// K6: selective scan.  Lane = (b,d); 16-state h in VGPRs; B/C loads are
// wave-uniform (same batch across the 32 lanes) -> single-line broadcasts.
// Only tokens >= SEG0 emit y (everything downstream needs just that segment).
// ---------------------------------------------------------------------------
__global__ void k_scan(const float* __restrict__ dtb,
                       const float* __restrict__ xa,
                       const float* __restrict__ Bb,
                       const float* __restrict__ Cb,
                       const float* __restrict__ A_log,
                       float* __restrict__ yseg) {
  const int b = blockIdx.x;     // 32
  const int d = threadIdx.x;    // 256
  float as[DST], h[DST];
#pragma unroll
  for (int s = 0; s < DST; s++) { as[s] = -__expf(A_log[d * DST + s]); h[s] = 0.f; }
  for (int t = 0; t < LRNA; t++) {
    const long base = (long)b * LRNA + t;
    const float dt = dtb[base * DIN + d];
    const float dtx = dt * xa[base * DIN + d];
    const float* Bp = Bb + base * DST;
    const float* Cp = Cb + base * DST;
    float y = 0.f;
#pragma unroll
    for (int s = 0; s < DST; s++) {
      float dA = __expf(dt * as[s]);
      h[s] = dA * h[s] + dtx * Bp[s];
      y += h[s] * Cp[s];
    }
    if (t >= SEG0) yseg[((long)b * SEGL + (t - SEG0)) * DIN + d] = y;
  }
}

// ---------------------------------------------------------------------------
// K7a: yg = (y + xa*D) * SiLU(z)  on the 134-token segment.
// ---------------------------------------------------------------------------
__global__ void k_gate(const float* __restrict__ yseg,
                       const float* __restrict__ xa,
                       const float* __restrict__ zseg,
                       const float* __restrict__ Dv,
                       float* __restrict__ yg) {
  const long i = (long)blockIdx.x * 256 + threadIdx.x;   // 32*134*256 exact
  const int d = (int)(i & 255);
  const long r = i >> 8;
  const int ts = (int)(r % SEGL);
  const int b = (int)(r / SEGL);
  const float z = zseg[i];
  const float xv = xa[((long)b * LRNA + SEG0 + ts) * DIN + d];
  yg[i] = (yseg[i] + xv * Dv[d]) * siluf(z);
}

// ---------------------------------------------------------------------------
// K7c: layernorm each of the 134 rows, running max over rows -> c1[b,128].
// ---------------------------------------------------------------------------
__global__ void k_ln_max(const float* __restrict__ f1seg,
                         const float* __restrict__ g,
                         const float* __restrict__ bb,
                         float* __restrict__ c1w) {
  __shared__ float red[EMB];
  const int b = blockIdx.x, e = threadIdx.x;  // 128 threads
  float mx = -3.4e38f;
  for (int ts = 0; ts < SEGL; ts++) {
    float v = f1seg[((long)b * SEGL + ts) * EMB + e];
    red[e] = v; __syncthreads();
    for (int off = 64; off > 0; off >>= 1) { if (e < off) red[e] += red[e + off]; __syncthreads(); }
    float mu = red[0] * (1.f / EMB); __syncthreads();
    float dv = v - mu;
    red[e] = dv * dv; __syncthreads();
    for (int off = 64; off > 0; off >>= 1) { if (e < off) red[e] += red[e + off]; __syncthreads(); }
    float var = red[0] * (1.f / EMB); __syncthreads();
    float nv = dv * rsqrtf(var + 1e-5f) * g[e] + bb[e];
    mx = fmaxf(mx, nv);
  }
  c1w[b * EMB + e] = mx;
}

// ---------------------------------------------------------------------------
// K8: entire ADT encoder (L=15) in one block per batch; all state in LDS.
// Only token 14 feeds out_proj/LN (c2 = f2[:,14,:]).
// ---------------------------------------------------------------------------
__global__ void k_adt(const float* __restrict__ p2,
                      const float* __restrict__ tok_w, const float* __restrict__ tok_b,
                      const float* __restrict__ emb_w, const float* __restrict__ emb_b,
                      const float* __restrict__ cls,   const float* __restrict__ pos,
                      const float* __restrict__ inw,   const float* __restrict__ convw,
                      const float* __restrict__ convb, const float* __restrict__ xw,
                      const float* __restrict__ dtw,   const float* __restrict__ dtbias,
                      const float* __restrict__ A_log, const float* __restrict__ Dv,
                      const float* __restrict__ outw,  const float* __restrict__ lng,
                      const float* __restrict__ lnb,   float* __restrict__ c2w) {
  __shared__ float t0s[14];
  __shared__ float seqs[15][EMB];
  __shared__ float xis[15][DIN];
  __shared__ float xas[15][DIN];
  __shared__ float dts[15][DIN];
  __shared__ float Bs[15][DST], Cs[15][DST];
  __shared__ float dbls[15][40];
  __shared__ float zlast[DIN];
  __shared__ float ys[DIN];
  __shared__ float f2row[EMB];
  __shared__ float red[EMB];
  const int b = blockIdx.x;
  const int tid = threadIdx.x;  // 256

  if (tid < 14) {
    float a = tok_b[tid];
    for (int k = 0; k < 14; k++) a += p2[b * 14 + k] * tok_w[tid * 14 + k];
    t0s[tid] = a;
  }
  __syncthreads();
  for (int i = tid; i < 15 * EMB; i += 256) {
    int c = i >> 7, e = i & 127;
    seqs[c][e] = (c == 0) ? cls[e]
                          : t0s[c - 1] * emb_w[e] + emb_b[e] + pos[(c - 1) * EMB + e];
  }
  __syncthreads();
  for (int i = tid; i < 15 * DIN; i += 256) {       // xi half of in_proj
    int c = i >> 8, n = i & 255;
    float a = 0.f;
    const float* w = inw + n * EMB;
    for (int k = 0; k < EMB; k++) a += seqs[c][k] * w[k];
    xis[c][n] = a;
  }
  {                                                  // z half, token 14 only
    float a = 0.f;
    const float* w = inw + (DIN + tid) * EMB;
    for (int k = 0; k < EMB; k++) a += seqs[14][k] * w[k];
    zlast[tid] = a;
  }
  __syncthreads();
  {                                                  // conv + SiLU
    const int d = tid;
    float w0 = convw[d * 4], w1 = convw[d * 4 + 1],
          w2 = convw[d * 4 + 2], w3 = convw[d * 4 + 3], cb = convb[d];
    float x0 = 0.f, x1 = 0.f, x2 = 0.f, x3 = 0.f;
    for (int t = 0; t < 15; t++) {
      x0 = x1; x1 = x2; x2 = x3; x3 = xis[t][d];
      xas[t][d] = siluf(w0 * x0 + w1 * x1 + w2 * x2 + w3 * x3 + cb);
    }
  }
  __syncthreads();
  for (int i = tid; i < 15 * 40; i += 256) {         // x_proj
    int c = i / 40, n = i % 40;
    float a = 0.f;
    const float* w = xw + n * DIN;
    for (int k = 0; k < DIN; k++) a += xas[c][k] * w[k];
    dbls[c][n] = a;
    if (n >= 8 && n < 24) Bs[c][n - 8] = a;
    else if (n >= 24)     Cs[c][n - 24] = a;
  }
  __syncthreads();
  for (int i = tid; i < 15 * DIN; i += 256) {        // dt_proj + softplus
    int c = i >> 8, d = i & 255;
    float s = dtbias[d];
#pragma unroll
    for (int r = 0; r < 8; r++) s += dbls[c][r] * dtw[d * 8 + r];
    dts[c][d] = softplusf(s);
  }
  __syncthreads();
  {                                                  // scan, keep y at t=14
    const int d = tid;
    float as[DST], h[DST];
#pragma unroll
    for (int s = 0; s < DST; s++) { as[s] = -__expf(A_log[d * DST + s]); h[s] = 0.f; }
    float ylast = 0.f;
    for (int t = 0; t < 15; t++) {
      float dt = dts[t][d];
      float dtx = dt * xas[t][d];
      float y = 0.f;
#pragma unroll
      for (int s = 0; s < DST; s++) {
        h[s] = __expf(dt * as[s]) * h[s] + dtx * Bs[t][s];
        y += h[s] * Cs[t][s];
      }
      ylast = y;
    }
    float z = zlast[d];
    ys[d] = (ylast + xas[14][d] * Dv[d]) * siluf(z);
  }
  __syncthreads();
  if (tid < EMB) {                                   // out_proj, token 14
    float a = 0.f;
    const float* w = outw + tid * DIN;
    for (int k = 0; k < DIN; k++) a += ys[k] * w[k];
    f2row[tid] = a;
  }
  __syncthreads();
  if (tid < EMB) red[tid] = f2row[tid];
  __syncthreads();
  for (int off = 64; off > 0; off >>= 1) { if (tid < off) red[tid] += red[tid + off]; __syncthreads(); }
  float mu = red[0] * (1.f / EMB); __syncthreads();
  if (tid < EMB) { float dv = f2row[tid] - mu; red[tid] = dv * dv; }
  __syncthreads();
  for (int off = 64; off > 0; off >>= 1) { if (tid < off) red[tid] += red[tid + off]; __syncthreads(); }
  float var = red[0] * (1.f / EMB); __syncthreads();
  if (tid < EMB)
    c2w[b * EMB + tid] = (f2row[tid] - mu) * rsqrtf(var + 1e-5f) * lng[tid] + lnb[tid];
}

// ---------------------------------------------------------------------------
// K9: fusion head.  One block (128 threads) per batch.
// ---------------------------------------------------------------------------
__device__ __forceinline__ float block_mean128(float v, float* red, int e) {
  red[e] = v; __syncthreads();
  for (int off = 64; off > 0; off >>= 1) { if (e < off) red[e] += red[e + off]; __syncthreads(); }
  float s = red[0]; __syncthreads();
  return s * (1.f / 128.f);
}

__global__ void k_fusion(const float* __restrict__ c1w, const float* __restrict__ c2w,
                         const float* __restrict__ ldc_rna_w, const float* __restrict__ ldc_rna_b,
                         const float* __restrict__ ldc_adt_w, const float* __restrict__ ldc_adt_b,
                         const float* __restrict__ ln1g, const float* __restrict__ ln1b,
                         const float* __restrict__ ln2g, const float* __restrict__ ln2b,
                         const float* __restrict__ lin1w, const float* __restrict__ lin1b,
                         const float* __restrict__ lin2w, const float* __restrict__ lin2b,
                         const float* __restrict__ ecaw,
                         const float* __restrict__ headw, const float* __restrict__ headb,
                         float* __restrict__ out) {
  __shared__ float c1s[EMB], c2s[EMB], n1[EMB], n2[EMB], o1s[EMB], o2s[EMB],
                   cf[EMB], fin[EMB], red[EMB];
  const int b = blockIdx.x, e = threadIdx.x;  // 128
  const float c1v = c1w[b * EMB + e];
  const float c2v = c2w[b * EMB + e];
  c1s[e] = c1v; c2s[e] = c2v;
  float md = block_mean128(c1v - c2v, red, e);
  float ap1 = sigf(md), ap2 = sigf(-md);
  float l1 = ldc_rna_b[e], l2 = ldc_adt_b[e];
  for (int i = 0; i < EMB; i++) {
    l1 += c1s[i] * ldc_rna_w[(e * EMB + i) * 3 + 1];
    l2 += c2s[i] * ldc_adt_w[(e * EMB + i) * 3 + 1];
  }
  float cross = c1v + c2v;
  float cc1 = ap1 * cross + l1 + c1v;
  float cc2 = ap2 * cross + l2 + c2v;
  // LN(cc1) -> n1 ; LN(cc2) -> n2
  {
    float mu = block_mean128(cc1, red, e);
    float dv = cc1 - mu;
    float var = block_mean128(dv * dv, red, e);
    n1[e] = dv * rsqrtf(var + 1e-5f) * ln1g[e] + ln1b[e];
  }
  {
    float mu = block_mean128(cc2, red, e);
    float dv = cc2 - mu;
    float var = block_mean128(dv * dv, red, e);
    n2[e] = dv * rsqrtf(var + 1e-5f) * ln2g[e] + ln2b[e];
  }
  __syncthreads();
  float o1v = lin1b[e], o2v = lin2b[e];
  for (int i = 0; i < EMB; i++) {
    o1v += n1[i] * lin1w[e * EMB + i];
    o2v += n2[i] * lin2w[e * EMB + i];
  }
  o1s[e] = o1v; o2s[e] = o2v;
  float cfv = o1v * o2v + o1v + o2v;
  cf[e] = cfv;
  __syncthreads();
  n1[e] = cfv * o1v;   // reuse as GEMV sources
  n2[e] = cfv * o2v;
  __syncthreads();
  float t1 = lin1b[e], t2 = lin2b[e];
  for (int i = 0; i < EMB; i++) {
    t1 += n1[i] * lin1w[e * EMB + i];
    t2 += n2[i] * lin2w[e * EMB + i];
  }
  float ew = ecaw[0];
  float m1 = block_mean128(t1, red, e);
  float m2 = block_mean128(t2, red, e);
  float x1 = t1 * sigf(m1 * ew) + o1v + cfv;
  float x2 = t2 * sigf(m2 * ew) + o2v + cfv;
  float fv = 0.5f * (x1 + x2);
  fin[e] = fv;
  __syncthreads();
  out[576 + b * EMB + e] = fv;                // final
  out[576 + 4096 + b * EMB + e] = o1v;        // o1
  out[576 + 8192 + b * EMB + e] = o2v;        // o2
  if (e < 18) {
    float a = headb[e];
    for (int i = 0; i < EMB; i++) a += fin[i] * headw[e * EMB + i];
    out[b * 18 + e] = a;                      // logits
  }
}

// ---------------------------------------------------------------------------
// Launcher.  Input order: patches1, patches2, then params flattened with
// jax.tree_leaves semantics (dict keys sorted alphabetically at each level).
// ---------------------------------------------------------------------------
extern "C" void kernel_launch(void* const* d_in, const int* in_sizes, int n_in,
                              void* d_out, int out_size, void* d_ws, size_t ws_size,
                              hipStream_t stream) {
  (void)in_sizes; (void)n_in; (void)out_size; (void)ws_size;
  const float* patches1  = (const float*)d_in[0];
  const float* patches2  = (const float*)d_in[1];
  const float* adt_cls   = (const float*)d_in[2];
  const float* adt_emb_b = (const float*)d_in[3];
  const float* adt_emb_w = (const float*)d_in[4];
  const float* adt_ln_b  = (const float*)d_in[5];
  const float* adt_ln_g  = (const float*)d_in[6];
  const float* adt_A_log = (const float*)d_in[7];
  const float* adt_D     = (const float*)d_in[8];
  const float* adt_conv_b= (const float*)d_in[9];
  const float* adt_conv_w= (const float*)d_in[10];
  const float* adt_dt_b  = (const float*)d_in[11];
  const float* adt_dt_w  = (const float*)d_in[12];
  const float* adt_in_w  = (const float*)d_in[13];
  const float* adt_out_w = (const float*)d_in[14];
  const float* adt_x_w   = (const float*)d_in[15];
  const float* adt_pos   = (const float*)d_in[16];
  const float* adt_tok_b = (const float*)d_in[17];
  const float* adt_tok_w = (const float*)d_in[18];
  const float* eca_w     = (const float*)d_in[19];
  const float* head_b    = (const float*)d_in[20];
  const float* head_w    = (const float*)d_in[21];
  const float* ldc_adt_b = (const float*)d_in[22];
  const float* ldc_adt_w = (const float*)d_in[23];
  const float* ldc_rna_b = (const float*)d_in[24];
  const float* ldc_rna_w = (const float*)d_in[25];
  const float* lin1_b    = (const float*)d_in[26];
  const float* lin1_w    = (const float*)d_in[27];
  const float* lin2_b    = (const float*)d_in[28];
  const float* lin2_w    = (const float*)d_in[29];
  const float* ln1_b     = (const float*)d_in[30];
  const float* ln1_g     = (const float*)d_in[31];
  const float* ln2_b     = (const float*)d_in[32];
  const float* ln2_g     = (const float*)d_in[33];
  const float* rna_cls   = (const float*)d_in[34];
  const float* rna_emb_b = (const float*)d_in[35];
  const float* rna_emb_w = (const float*)d_in[36];
  const float* rna_ln_b  = (const float*)d_in[37];
  const float* rna_ln_g  = (const float*)d_in[38];
  const float* rna_A_log = (const float*)d_in[39];
  const float* rna_D     = (const float*)d_in[40];
  const float* rna_conv_b= (const float*)d_in[41];
  const float* rna_conv_w= (const float*)d_in[42];
  const float* rna_dt_b  = (const float*)d_in[43];
  const float* rna_dt_w  = (const float*)d_in[44];
  const float* rna_in_w  = (const float*)d_in[45];
  const float* rna_out_w = (const float*)d_in[46];
  const float* rna_x_w   = (const float*)d_in[47];
  const float* rna_pos   = (const float*)d_in[48];
  const float* rna_tok_b = (const float*)d_in[49];
  const float* rna_tok_w = (const float*)d_in[50];

  // workspace layout (floats)
  float* ws = (float*)d_ws;
  float* t0    = ws; ws += (size_t)BATCH * NTOK;
  float* seq   = ws; ws += (size_t)BATCH * LRNA * EMB;
  float* xa    = ws; ws += (size_t)BATCH * LRNA * DIN;
  float* dtb   = ws; ws += (size_t)BATCH * LRNA * DIN;
  float* Bb    = ws; ws += (size_t)BATCH * LRNA * DST;
  float* Cb    = ws; ws += (size_t)BATCH * LRNA * DST;
  float* zseg  = ws; ws += (size_t)BATCH * SEGL * DIN;
  float* yseg  = ws; ws += (size_t)BATCH * SEGL * DIN;
  float* yg    = ws; ws += (size_t)BATCH * SEGL * DIN;
  float* f1seg = ws; ws += (size_t)BATCH * SEGL * EMB;
  float* c1w   = ws; ws += (size_t)BATCH * EMB;
  float* c2w   = ws; ws += (size_t)BATCH * EMB;

  const int M_full = BATCH * LRNA;   // 64032 = 16 * 4002
  const int M_seg  = BATCH * SEGL;   // 4288  = 16 * 268

  k_tokens_gemm<<<63, 256, 0, stream>>>(patches1, rna_tok_w, rna_tok_b, t0);
  k_embed<<<dim3(LRNA, BATCH), EMB, 0, stream>>>(t0, rna_emb_w, rna_emb_b,
                                                 rna_pos, rna_cls, seq);
  // in_proj (xi half) over all tokens: C[64032,256] = seq @ W[0:256,:]^T
  k_wmma_gemm<<<dim3(1001, 16), 128, 0, stream>>>(seq, rna_in_w, xa,
                                                  M_full, DIN, EMB,
                                                  LRNA, LRNA, 0);
  k_conv_silu<<<BATCH, DIN, 0, stream>>>(xa, rna_conv_w, rna_conv_b);
  k_xproj_dt<<<M_full, DIN, 0, stream>>>(xa, rna_x_w, rna_dt_w, rna_dt_b,
                                         Bb, Cb, dtb);
  // z half of in_proj, only tokens 1867..2000 (all that's consumed downstream)
  k_wmma_gemm<<<dim3(67, 16), 128, 0, stream>>>(seq, rna_in_w + DIN * EMB, zseg,
                                                M_seg, DIN, EMB,
                                                SEGL, LRNA, SEG0);
  k_scan<<<BATCH, DIN, 0, stream>>>(dtb, xa, Bb, Cb, rna_A_log, yseg);
  k_gate<<<M_seg, 256, 0, stream>>>(yseg, xa, zseg, rna_D, yg);
  // out_proj on the segment: C[4288,128] = yg @ W[128,256]^T
  k_wmma_gemm<<<dim3(67, 8), 128, 0, stream>>>(yg, rna_out_w, f1seg,
                                               M_seg, EMB, DIN,
                                               M_seg, M_seg, 0);
  k_ln_max<<<BATCH, EMB, 0, stream>>>(f1seg, rna_ln_g, rna_ln_b, c1w);
  k_adt<<<BATCH, 256, 0, stream>>>(patches2, adt_tok_w, adt_tok_b,
                                   adt_emb_w, adt_emb_b, adt_cls, adt_pos,
                                   adt_in_w, adt_conv_w, adt_conv_b,
                                   adt_x_w, adt_dt_w, adt_dt_b,
                                   adt_A_log, adt_D, adt_out_w,
                                   adt_ln_g, adt_ln_b, c2w);
  k_fusion<<<BATCH, EMB, 0, stream>>>(c1w, c2w,
                                      ldc_rna_w, ldc_rna_b, ldc_adt_w, ldc_adt_b,
                                      ln1_g, ln1_b, ln2_g, ln2_b,
                                      lin1_w, lin1_b, lin2_w, lin2_b,
                                      eca_w, head_w, head_b, (float*)d_out);
}